// VonMisesFisherLoss_43164421325213
// MI455X (gfx1250) — compile-verified
//
#include <hip/hip_runtime.h>
#include <hip/hip_bf16.h>
#include <stdint.h>

typedef unsigned int u32x4 __attribute__((ext_vector_type(4)));
typedef int          i32x4 __attribute__((ext_vector_type(4)));
typedef int          i32x8 __attribute__((ext_vector_type(8)));

#define TILE_ROWS     1024
#define TILE_FLOATS   (TILE_ROWS * 3)      // 3072 floats = 12 KB
#define BLOCK_THREADS 256
#define MAX_BLOCKS    2048

#if defined(__AMDGCN__) && __has_builtin(__builtin_amdgcn_tensor_load_to_lds) && __has_builtin(__builtin_amdgcn_s_wait_tensorcnt)
#define USE_TDM 1
#else
#define USE_TDM 0
#endif

// ---- per-row vMF loss element (m=3 closed forms) -----------------------
// exact : logC = log k - log sinh k - log(4*pi)
//         log sinh k = k + log1p(-exp(-2k)) - log 2
// approx: logC = -sqrt(4 + k^2) - offset, offset = approx(100)-exact(100) = -2.7872911
__device__ __forceinline__ float vmf_element(float p0, float p1, float p2,
                                             float t0, float t1, float t2) {
  float k2  = fmaf(p0, p0, fmaf(p1, p1, p2 * p2));
  float k   = __builtin_sqrtf(k2);
  float dot = fmaf(p0, t0, fmaf(p1, t1, p2 * t2));
  float e2k = __expf(-2.0f * k);
  float logsinh = k + log1pf(-e2k) - 0.69314718056f;       // - ln 2
  float exact   = __logf(k) - logsinh - 2.53102424697f;    // - ln(4*pi)
  float far     = 2.7872911f - __builtin_sqrtf(4.0f + k2); // approx - offset
  float logc    = (k < 100.0f) ? exact : far;
  return -logc - dot;
}

#if USE_TDM
// Issue one 1-D TDM load: nfloats contiguous f32 from gsrc -> LDS byte offset.
// D# per CDNA5 ISA ch.8: group0 = {flags, lds_addr, global_addr, type=2},
// group1 = {mask/data_size, dims, tile dims, strides}; groups 2/3 zero (<=2D).
// amdgpu-toolchain (clang-23) builtin arity: 6 args
//   (u32x4 g0, i32x8 g1, i32x4, i32x4, i32x8, i32 cpol)
__device__ __forceinline__ void tdm_load_1d(uint32_t lds_byte_off,
                                            const float* gsrc,
                                            uint32_t nfloats) {
  uint64_t ga = (uint64_t)(uintptr_t)gsrc;
  u32x4 g0;
  g0[0] = 1u;                                                // count=1 (valid), user mode
  g0[1] = lds_byte_off;                                      // lds_addr (bytes)
  g0[2] = (uint32_t)ga;                                      // global_addr[31:0]
  g0[3] = (uint32_t)((ga >> 32) & 0x01FFFFFFu) | (2u << 30); // addr[56:32] | type=2
  i32x8 g1;
  g1[0] = (int)(2u << 16);                          // wg_mask=0, data_size=2 (4B)
  g1[1] = (int)((nfloats & 0xFFFFu) << 16);         // tensor_dim0[15:0] @ bits 63:48
  g1[2] = (int)((nfloats >> 16) | (1u << 16));      // tensor_dim0[31:16] | tensor_dim1=1
  g1[3] = (int)((nfloats & 0xFFFFu) << 16);         // tile_dim0 = nfloats (<=65535)
  g1[4] = 0;                                        // tile_dim1=0, tile_dim2=0 (1-D)
  g1[5] = (int)nfloats;                             // tensor_dim0_stride[31:0]
  g1[6] = 0;
  g1[7] = 0;
  i32x4 z4 = {0, 0, 0, 0};
  i32x8 z8 = {0, 0, 0, 0, 0, 0, 0, 0};
  __builtin_amdgcn_tensor_load_to_lds(g0, g1, z4, z4, z8, 0);
}
#endif

__global__ __launch_bounds__(BLOCK_THREADS)
void vmf_partial_kernel(const float* __restrict__ pred,
                        const float* __restrict__ targ,
                        double* __restrict__ partials, int numRows) {
  __shared__ float sh[2][2][TILE_FLOATS]; // [buf][pred/targ][floats] = 48 KB
  const int tidx     = threadIdx.x;
  const int numTiles = (numRows + TILE_ROWS - 1) / TILE_ROWS;
  const int stride   = gridDim.x;
  float lsum = 0.0f;

#if USE_TDM
  const uint32_t totalFloats = (uint32_t)numRows * 3u;
  auto issue = [&](int t, int b) {
    if (tidx == 0) { // one TDM pair per tile, issued by wave 0 (EXEC ignored by TDM)
      uint32_t base = (uint32_t)t * (uint32_t)TILE_FLOATS;
      uint32_t nf   = totalFloats - base;
      if (nf > (uint32_t)TILE_FLOATS) nf = (uint32_t)TILE_FLOATS;
      tdm_load_1d((uint32_t)(uintptr_t)&sh[b][0][0], pred + base, nf);
      tdm_load_1d((uint32_t)(uintptr_t)&sh[b][1][0], targ + base, nf);
    }
  };

  int tile = blockIdx.x;
  int buf  = 0;
  if (tile < numTiles) issue(tile, 0);
  for (; tile < numTiles; tile += stride) {
    int  nxt      = tile + stride;
    bool havenext = nxt < numTiles;
    if (havenext) issue(nxt, buf ^ 1);           // prefetch into other buffer
    if (havenext) __builtin_amdgcn_s_wait_tensorcnt(2); // current 2 done, next 2 in flight
    else          __builtin_amdgcn_s_wait_tensorcnt(0);
    __syncthreads();                              // LDS tile visible to all waves

    int rows = numRows - tile * TILE_ROWS;
    if (rows > TILE_ROWS) rows = TILE_ROWS;
#pragma unroll
    for (int w = 0; w < TILE_ROWS / BLOCK_THREADS; ++w) {
      int r = tidx + w * BLOCK_THREADS;
      if (r < rows) {
        const float* P = &sh[buf][0][3 * r];
        const float* T = &sh[buf][1][3 * r];
        lsum += vmf_element(P[0], P[1], P[2], T[0], T[1], T[2]);
      }
    }
    __syncthreads(); // all waves done reading buf before it is refilled
    buf ^= 1;
  }
#else
  (void)sh;
  for (long long r = (long long)blockIdx.x * BLOCK_THREADS + tidx; r < numRows;
       r += (long long)gridDim.x * BLOCK_THREADS) {
    const float* P = pred + 3 * r;
    const float* T = targ + 3 * r;
    lsum += vmf_element(P[0], P[1], P[2], T[0], T[1], T[2]);
  }
#endif

  // wave32 tree reduce (f32), then per-block double partial
#pragma unroll
  for (int off = 16; off > 0; off >>= 1) lsum += __shfl_xor(lsum, off, 32);
  __shared__ float wsum[BLOCK_THREADS / 32];
  int wid = tidx >> 5, lane = tidx & 31;
  if (lane == 0) wsum[wid] = lsum;
  __syncthreads();
  if (tidx == 0) {
    double b = 0.0;
#pragma unroll
    for (int i = 0; i < BLOCK_THREADS / 32; ++i) b += (double)wsum[i];
    partials[blockIdx.x] = b;
  }
}

__global__ __launch_bounds__(256)
void vmf_final_kernel(const double* __restrict__ partials, int nParts,
                      float* __restrict__ out, int numRows) {
  double s = 0.0;
  for (int i = threadIdx.x; i < nParts; i += 256) s += partials[i];
#pragma unroll
  for (int off = 16; off > 0; off >>= 1) s += __shfl_xor(s, off, 32);
  __shared__ double ws[8];
  int wid = threadIdx.x >> 5, lane = threadIdx.x & 31;
  if (lane == 0) ws[wid] = s;
  __syncthreads();
  if (threadIdx.x == 0) {
    double t = 0.0;
#pragma unroll
    for (int i = 0; i < 8; ++i) t += ws[i];
    out[0] = (float)(t / (double)numRows);
  }
}

extern "C" void kernel_launch(void* const* d_in, const int* in_sizes, int n_in,
                              void* d_out, int out_size, void* d_ws, size_t ws_size,
                              hipStream_t stream) {
  const float* pred = (const float*)d_in[0];
  const float* targ = (const float*)d_in[1];
  int numRows  = in_sizes[0] / 3;
  int numTiles = (numRows + TILE_ROWS - 1) / TILE_ROWS;
  int blocks   = numTiles < MAX_BLOCKS ? numTiles : MAX_BLOCKS;
  if (blocks < 1) blocks = 1;
  double* partials = (double*)d_ws;

  vmf_partial_kernel<<<blocks, BLOCK_THREADS, 0, stream>>>(pred, targ, partials, numRows);
  vmf_final_kernel<<<1, 256, 0, stream>>>(partials, blocks, (float*)d_out, numRows);
}